// Attns_61443802137046
// MI455X (gfx1250) — compile-verified
//
#include <hip/hip_runtime.h>

typedef __attribute__((ext_vector_type(2))) float v2f;
typedef __attribute__((ext_vector_type(4))) float v4f;
typedef __attribute__((ext_vector_type(8))) float v8f;

constexpr int Bn = 32, Cn = 256, Hn = 128, Wn = 128;
constexpr int HWn  = Hn * Wn;          // 16384
constexpr int NPIX = Bn * HWn;         // 524288
constexpr long long NTOT = (long long)Bn * Cn * HWn;   // 134217728

// ---------------------------------------------------------------------------
// Kernel 1: channel-wise mean & max over C. Streams y (512 MiB) exactly once
// with non-temporal b128 loads; writes the 4 MiB s-planes (stay hot in L2).
// ---------------------------------------------------------------------------
__global__ __launch_bounds__(256) void reduce_y_kernel(
        const float* __restrict__ y,
        float* __restrict__ savg,
        float* __restrict__ smax) {
    int p4   = blockIdx.x * blockDim.x + threadIdx.x;   // one float4 of pixels
    int flat = p4 * 4;
    int b    = flat / HWn;
    int off  = flat - b * HWn;
    const v4f* yb = (const v4f*)(y + (size_t)b * Cn * HWn + off);

    v4f s = __builtin_nontemporal_load(yb);
    v4f m = s;
    #pragma unroll 8
    for (int c = 1; c < Cn; ++c) {
        v4f t = __builtin_nontemporal_load(yb + (size_t)c * (HWn / 4));
        s += t;
        m.x = fmaxf(m.x, t.x); m.y = fmaxf(m.y, t.y);
        m.z = fmaxf(m.z, t.z); m.w = fmaxf(m.w, t.w);
    }
    s *= (1.0f / Cn);
    *(v4f*)(savg + flat) = s;
    *(v4f*)(smax + flat) = m;
}

// ---------------------------------------------------------------------------
// Kernel 2: 3x3 conv (2 ch -> 1) + sigmoid via V_WMMA_F32_16X16X4_F32.
// One wave handles 16 consecutive pixels (W=128 is a multiple of 16, so no
// row wrap inside a wave). K-dim = 18 taps (2ch x 3x3), padded to 20 ->
// 5 chained WMMAs. B-matrix = weights broadcast across all 16 columns, so
// every column of D equals the 16 gate pre-activations.
// A 16x4 layout: lane L holds M = L%16, K = 2*(L>>4) + {0,1} (v0,v1).
// D 16x16 layout: lane L, reg v holds D[v + 8*(L>>4)][L%16].
// ---------------------------------------------------------------------------
__device__ __forceinline__ float tap_s(const float* __restrict__ savg,
                                       const float* __restrict__ smax,
                                       int b, int h, int w, int t) {
    if (t >= 18) return 0.0f;                 // K padding 18 -> 20
    int ch = t / 9;
    int r  = t - ch * 9;
    int hh = h + r / 3 - 1;
    int ww = w + r % 3 - 1;
    if (hh < 0 || hh >= Hn || ww < 0 || ww >= Wn) return 0.0f;  // zero pad
    const float* p = ch ? smax : savg;
    return p[(size_t)b * HWn + hh * Wn + ww];
}

__global__ __launch_bounds__(256) void gate_wmma_kernel(
        const float* __restrict__ savg,
        const float* __restrict__ smax,
        const float* __restrict__ conv_w,   // [1,2,3,3] -> 18 floats, tap order
        float* __restrict__ gate) {
    int gtid    = blockIdx.x * blockDim.x + threadIdx.x;
    int wave    = gtid >> 5;
    int lane    = threadIdx.x & 31;
    int m       = lane & 15;          // pixel row M of A
    int hi      = lane >> 4;          // which K-pair this lane holds
    int pixBase = wave * 16;
    int pix     = pixBase + m;
    int b       = pix / HWn;
    int rem     = pix - b * HWn;
    int h       = rem / Wn;
    int w       = rem - h * Wn;

    v8f acc = {};                     // C starts at zero
    #pragma unroll
    for (int kk = 0; kk < 5; ++kk) {
        int t0 = kk * 4 + hi * 2;
        v2f a, bm;
        a.x  = tap_s(savg, smax, b, h, w, t0);
        a.y  = tap_s(savg, smax, b, h, w, t0 + 1);
        bm.x = (t0     < 18) ? conv_w[t0]     : 0.0f;
        bm.y = (t0 + 1 < 18) ? conv_w[t0 + 1] : 0.0f;
        acc = __builtin_amdgcn_wmma_f32_16x16x4_f32(
                  false, a, false, bm, (short)0, acc, false, false);
    }

    // Columns of D are identical; lanes 0 and 16 each own 8 rows -> store
    // sigmoid(gate) for their 8 pixels (gate plane is tiny, stays in L2).
    if (m == 0) {
        float g[8];
        #pragma unroll
        for (int v = 0; v < 8; ++v)
            g[v] = 1.0f / (1.0f + __expf(-acc[v]));
        float* dst = gate + pixBase + hi * 8;
        v4f g0 = {g[0], g[1], g[2], g[3]};
        v4f g1 = {g[4], g[5], g[6], g[7]};
        *(v4f*)(dst)     = g0;
        *(v4f*)(dst + 4) = g1;
    }
}

// ---------------------------------------------------------------------------
// Kernel 3: out = x * gate (broadcast over C). Streams x and out (1 GiB)
// with non-temporal float4 traffic; gate reads hit L2.
// ---------------------------------------------------------------------------
__global__ __launch_bounds__(256) void apply_kernel(
        const float* __restrict__ x,
        const float* __restrict__ gate,
        float* __restrict__ out) {
    size_t i4   = (size_t)blockIdx.x * blockDim.x + threadIdx.x;
    size_t flat = i4 * 4;
    size_t pix  = flat & (size_t)(HWn - 1);          // flat % HW
    size_t b    = flat / ((size_t)Cn * HWn);
    v4f xv = __builtin_nontemporal_load((const v4f*)(x + flat));
    v4f g  = *(const v4f*)(gate + b * HWn + pix);
    v4f o  = xv * g;
    __builtin_nontemporal_store(o, (v4f*)(out + flat));
}

// ---------------------------------------------------------------------------
extern "C" void kernel_launch(void* const* d_in, const int* in_sizes, int n_in,
                              void* d_out, int out_size, void* d_ws, size_t ws_size,
                              hipStream_t stream) {
    const float* x      = (const float*)d_in[0];
    const float* y      = (const float*)d_in[1];
    const float* conv_w = (const float*)d_in[2];
    float* out = (float*)d_out;

    float* savg = (float*)d_ws;            // NPIX floats
    float* smax = savg + NPIX;             // NPIX floats
    float* gate = smax + NPIX;             // NPIX floats  (6 MiB total)

    // 1) channel reduce: 131072 float4-threads
    reduce_y_kernel<<<(NPIX / 4) / 256, 256, 0, stream>>>(y, savg, smax);
    // 2) WMMA conv + sigmoid: 32768 waves x 16 pixels, 8 waves/block
    gate_wmma_kernel<<<(NPIX / 16) / 8, 256, 0, stream>>>(savg, smax, conv_w, gate);
    // 3) broadcast multiply: 33554432 float4-threads
    apply_kernel<<<(int)((NTOT / 4) / 256), 256, 0, stream>>>(x, gate, out);
}